// TestModel_7009386627779
// MI455X (gfx1250) — compile-verified
//
#include <hip/hip_runtime.h>
#include <hip/hip_bf16.h>

// out[h,b,j,k] = core[h] * sum_a q[a,j,h] * k[a,k,h] * v[a,b,h]
// H=16, A=16, J=K=B=192.  Per (h,b): GEMM  (J x A) * (A x K), A-contraction=16,
// done as 4 chained V_WMMA_F32_16X16X4_F32 per 16x16 tile.

#define H_DIM 16
#define A_DIM 16
#define JKB   192
#define LDS_STRIDE 200   // 192 + 8 pad: 2-row shift = 16 banks -> no conflicts

typedef __attribute__((ext_vector_type(2))) float v2f;
typedef __attribute__((ext_vector_type(8))) float v8f;

__global__ __launch_bounds__(256, 1)
void rel_attn_tri_kernel(const float* __restrict__ core,
                         const float* __restrict__ q,
                         const float* __restrict__ kmat,
                         const float* __restrict__ vmat,
                         float* __restrict__ out) {
    __shared__ float Qs[A_DIM * LDS_STRIDE];   // Qv[a][j] = q*v[a]*core
    __shared__ float Ks[A_DIM * LDS_STRIDE];   // K  [a][k]

    const int hb = blockIdx.x;
    const int h  = hb / JKB;          // consecutive blocks share h -> L2 reuse
    const int b  = hb - h * JKB;
    const int tid = threadIdx.x;

    const float ch = core[h];

    // ---- stage Qv and K into LDS (inputs are tiny; L2-resident) ----
    for (int idx = tid; idx < A_DIM * JKB; idx += 256) {
        const int a = idx / JKB;
        const int j = idx - a * JKB;
        // x[a,j,h] layout: a*J*H + j*H + h
        const float vb = vmat[a * (JKB * H_DIM) + b * H_DIM + h];
        Qs[a * LDS_STRIDE + j] = q   [a * (JKB * H_DIM) + j * H_DIM + h] * vb * ch;
        Ks[a * LDS_STRIDE + j] = kmat[a * (JKB * H_DIM) + j * H_DIM + h];
    }
    __syncthreads();

    const int lane = tid & 31;
    const int wave = tid >> 5;
    const int m    = lane & 15;           // M (A-frag) / N (B-frag) index
    const int hi   = lane >> 4;           // 0: lanes 0-15, 1: lanes 16-31
    const int hi2  = hi << 1;             // K-offset 0 or 2 inside a K=4 step

    float* const outbase = out + ((size_t)h * JKB + b) * (size_t)(JKB * JKB);

    // 144 tiles of 16x16 (12 x 12), 18 per wave
    for (int t = wave * 18; t < wave * 18 + 18; ++t) {
        const int jt = t / 12;
        const int kt = t - jt * 12;

        const float* Acol = &Qs[jt * 16 + m];   // column j = jt*16 + m
        const float* Bcol = &Ks[kt * 16 + m];   // column k = kt*16 + m

        v8f c = {};
#pragma unroll
        for (int a0 = 0; a0 < A_DIM; a0 += 4) {
            v2f av, bv;
            // ISA 16x4 f32 A layout: VGPR0 = K a0 (lanes0-15) / a0+2 (lanes16-31),
            //                        VGPR1 = K a0+1 / a0+3. B mirrors with N on lanes.
            av[0] = Acol[(a0 + hi2)     * LDS_STRIDE];
            av[1] = Acol[(a0 + hi2 + 1) * LDS_STRIDE];
            bv[0] = Bcol[(a0 + hi2)     * LDS_STRIDE];
            bv[1] = Bcol[(a0 + hi2 + 1) * LDS_STRIDE];
            c = __builtin_amdgcn_wmma_f32_16x16x4_f32(
                    /*neg_a=*/false, av, /*neg_b=*/false, bv,
                    /*c_mod=*/(short)0, c, /*reuse_a=*/false, /*reuse_b=*/false);
        }

        // C layout: VGPR r -> M = r (lanes 0-15) / r+8 (lanes 16-31), N = lane&15.
        const int jrow = jt * 16 + hi * 8;
        float* o = outbase + (size_t)jrow * JKB + kt * 16 + m;
#pragma unroll
        for (int r = 0; r < 8; ++r) {
            __builtin_nontemporal_store(c[r], o + (size_t)r * JKB);  // NT: 453MB stream
        }
    }
}

extern "C" void kernel_launch(void* const* d_in, const int* in_sizes, int n_in,
                              void* d_out, int out_size, void* d_ws, size_t ws_size,
                              hipStream_t stream) {
    const float* core = (const float*)d_in[0];   // core_value [16]
    const float* q    = (const float*)d_in[1];   // rw_head_q  [16,192,16]
    const float* k    = (const float*)d_in[2];   // w_head_k   [16,192,16]
    const float* v    = (const float*)d_in[3];   // w_head_v   [16,192,16]
    float* out        = (float*)d_out;           // [16,192,192,192]

    dim3 grid(H_DIM * JKB);   // 3072 blocks, one (h,b) pair each
    dim3 block(256);          // 8 wave32s
    hipLaunchKernelGGL(rel_attn_tri_kernel, grid, block, 0, stream,
                       core, q, k, v, out);
}